// HSIModual_68521908240953
// MI455X (gfx1250) — compile-verified
//
#include <hip/hip_runtime.h>
#include <math.h>

#define Bn 4
#define Ln 224
#define Hn 1024
#define NEGC 1.0e9f

typedef float v2f __attribute__((ext_vector_type(2)));
typedef float v8f __attribute__((ext_vector_type(8)));

__device__ __forceinline__ v8f wmma4(v2f a, v2f b, v8f c) {
  // D = A(16x4 f32) * B(4x16 f32) + C(16x16 f32)
  return __builtin_amdgcn_wmma_f32_16x16x4_f32(false, a, false, b, (short)0, c, false, false);
}

// ---------------- generic 16x16-tile fp32 WMMA GEMM (one wave per tile) ------
// C[bz] = A[bz]*B[bz] (+ A2*B2 if DUAL). BT: B addressed transposed. ZD: zero diag.
// K must be a multiple of 16. Branch-free inner loop: 8 loads then 4 WMMAs.
template<bool BT, bool ZD, bool DUAL>
__global__ __launch_bounds__(32) void k_gemm(
    const float* __restrict__ A, const float* __restrict__ Bm,
    const float* __restrict__ A2, const float* __restrict__ B2,
    float* __restrict__ C,
    int K, int lda, int ldb, int ldc,
    long sA, long sB, long sC)
{
  const int lane = threadIdx.x;          // 0..31 (wave32)
  const int tn = blockIdx.x * 16;
  const int tm = blockIdx.y * 16;
  const long bz = blockIdx.z;
  A  += bz * sA;  Bm += bz * sB;  C += bz * sC;

  const int mr = lane & 15;              // A row / B col / C col
  const int kq = (lane >> 4) << 1;       // 0 or 2 (K sub-offset per half-wave)
  const long ldbl = ldb;

  v8f acc = {0.f,0.f,0.f,0.f,0.f,0.f,0.f,0.f};

  {
    const float* ap = A + (long)(tm + mr) * lda + kq;          // v2f per step
    const float* bp = BT ? (Bm + (long)(tn + mr) * ldbl + kq)  // v2f per step
                         : (Bm + (long)kq * ldbl + tn + mr);   // 2 x b32 per step
    const int niter = K >> 4;
    #pragma unroll 2
    for (int it = 0; it < niter; ++it) {
      v2f a0 = *(const v2f*)(ap);
      v2f a1 = *(const v2f*)(ap + 4);
      v2f a2 = *(const v2f*)(ap + 8);
      v2f a3 = *(const v2f*)(ap + 12);
      v2f b0, b1, b2, b3;
      if (BT) {
        b0 = *(const v2f*)(bp);
        b1 = *(const v2f*)(bp + 4);
        b2 = *(const v2f*)(bp + 8);
        b3 = *(const v2f*)(bp + 12);
        bp += 16;
      } else {
        b0.x = bp[0];          b0.y = bp[ldbl];
        b1.x = bp[4 * ldbl];   b1.y = bp[5 * ldbl];
        b2.x = bp[8 * ldbl];   b2.y = bp[9 * ldbl];
        b3.x = bp[12 * ldbl];  b3.y = bp[13 * ldbl];
        bp += 16 * ldbl;
      }
      ap += 16;
      acc = wmma4(a0, b0, acc);
      acc = wmma4(a1, b1, acc);
      acc = wmma4(a2, b2, acc);
      acc = wmma4(a3, b3, acc);
    }
  }
  if (DUAL) {  // fused second GEMM pair (conv k=2)
    const float* ap = A2 + (long)(tm + mr) * lda + kq;
    const float* bp = B2 + (long)kq * ldbl + tn + mr;
    const int niter = K >> 4;
    #pragma unroll 2
    for (int it = 0; it < niter; ++it) {
      v2f a0 = *(const v2f*)(ap);
      v2f a1 = *(const v2f*)(ap + 4);
      v2f a2 = *(const v2f*)(ap + 8);
      v2f a3 = *(const v2f*)(ap + 12);
      v2f b0, b1, b2, b3;
      b0.x = bp[0];          b0.y = bp[ldbl];
      b1.x = bp[4 * ldbl];   b1.y = bp[5 * ldbl];
      b2.x = bp[8 * ldbl];   b2.y = bp[9 * ldbl];
      b3.x = bp[12 * ldbl];  b3.y = bp[13 * ldbl];
      bp += 16 * ldbl;
      ap += 16;
      acc = wmma4(a0, b0, acc);
      acc = wmma4(a1, b1, acc);
      acc = wmma4(a2, b2, acc);
      acc = wmma4(a3, b3, acc);
    }
  }
  const int rbase = tm + ((lane >> 4) << 3);
  const int c = tn + mr;
  #pragma unroll
  for (int v = 0; v < 8; ++v) {
    int r = rbase + v;
    float val = acc[v];
    if (ZD && r == c) val = 0.f;
    C[(long)r * ldc + c] = val;
  }
}

// ---------------- small utility kernels --------------------------------------
__global__ void k_init(unsigned* m) { *m = 0u; }

// conv_w[O,I,2] -> Wt0[I,O], Wt1[I,O]  (so GEMM B operand is K-major/coalesced)
__global__ void k_trans(const float* __restrict__ cw, float* __restrict__ wt0,
                        float* __restrict__ wt1) {
  int idx = blockIdx.x * blockDim.x + threadIdx.x;
  if (idx >= Hn * Hn) return;
  int i = idx / Hn, o = idx % Hn;                  // write-contiguous in o
  wt0[(long)i * Hn + o] = cw[((long)o * Hn + i) * 2 + 0];
  wt1[(long)i * Hn + o] = cw[((long)o * Hn + i) * 2 + 1];
}

// hshift[b,l,:] = h[b,l+1,:], zero at l = L-1
__global__ void k_shift(const float* __restrict__ h, float* __restrict__ hs) {
  long idx = (long)blockIdx.x * blockDim.x + threadIdx.x;
  if (idx >= (long)Bn * Ln * Hn) return;
  int l = (int)((idx / Hn) % Ln);
  hs[idx] = (l < Ln - 1) ? h[idx + Hn] : 0.f;
}

__device__ __forceinline__ float blk_sum(float v, float* red) {
  int tid = threadIdx.x;
  red[tid] = v; __syncthreads();
  for (int w = 128; w > 0; w >>= 1) { if (tid < w) red[tid] += red[tid + w]; __syncthreads(); }
  float r = red[0]; __syncthreads();
  return r;
}

// LayerNorm -> tanh -> dot(weight) -> mask fill. One 256-thread block per row.
__global__ void k_ln(const float* __restrict__ pre, const float* __restrict__ bias,
                     const float* __restrict__ wv, const float* __restrict__ wb,
                     const int* __restrict__ ids, float* __restrict__ outv,
                     int shiftMask, float fill)
{
  __shared__ float red[256];
  const int row = blockIdx.x, tid = threadIdx.x;
  const float* x = pre + (long)row * Hn;
  float s = 0.f;
  for (int c = tid; c < Hn; c += 256) s += x[c] + bias[c];
  float mean = blk_sum(s, red) * (1.f / Hn);
  float s2 = 0.f;
  for (int c = tid; c < Hn; c += 256) { float d = x[c] + bias[c] - mean; s2 += d * d; }
  float rstd = rsqrtf(blk_sum(s2, red) * (1.f / Hn) + 1e-5f);
  float acc = 0.f;
  for (int c = tid; c < Hn; c += 256)
    acc += tanhf((x[c] + bias[c] - mean) * rstd) * wv[c];
  float dot = blk_sum(acc, red);
  if (tid == 0) {
    int b = row / Ln, l = row % Ln;
    bool m = shiftMask ? ((l < Ln - 1) && ids[b * Ln + l + 1] != 0)
                       : (ids[b * Ln + l] != 0);
    outv[row] = m ? (dot + wb[0]) : fill;
  }
}

// margin = max over masked min(relu(d_pad - cummax_h), relu(d - cummax_h))
__global__ void k_margin(const float* __restrict__ height, const float* __restrict__ draw,
                         const int* __restrict__ ids, unsigned* __restrict__ mb)
{
  int t = blockIdx.x * blockDim.x + threadIdx.x;
  if (t >= Bn * Ln) return;
  int b = t / Ln, i = t % Ln;
  const float* hb = height + b * Ln;
  const float* db = draw + b * Ln;
  float dpad = (i == 0) ? NEGC : db[i - 1];
  float run = -INFINITY, loc = 0.f;
  for (int j = i; j < Ln; ++j) {
    run = fmaxf(run, hb[j]);
    float v = fminf(fmaxf(dpad - run, 0.f), fmaxf(db[j] - run, 0.f));
    bool mm = (i == 0) ? ((j < Ln - 1) && ids[b * Ln + j + 1] != 0)
                       : (ids[b * Ln + j] != 0);
    if (mm) loc = fmaxf(loc, v);
  }
  atomicMax(mb, __float_as_uint(loc));     // loc >= 0 -> uint order == float order
}

// per batch: final distance, e[h]=exp(s1*(height-Hmax)), prefix sums Pex, indexes
__global__ void k_finalize(const float* __restrict__ height, const float* __restrict__ draw,
                           const float* __restrict__ marginF, const float* __restrict__ scaler,
                           float* __restrict__ od, float* __restrict__ e,
                           float* __restrict__ pex, float* __restrict__ oidx)
{
  __shared__ float red[256];
  int b = blockIdx.x, tid = threadIdx.x;
  float margin = marginF[0];
  float s1 = expf(scaler[1]);
  for (int l = tid; l < Ln; l += 256) od[b * Ln + l] = draw[b * Ln + l] - margin;
  float m = -INFINITY;
  for (int l = tid; l < Ln; l += 256) m = fmaxf(m, height[b * Ln + l]);
  red[tid] = m; __syncthreads();
  for (int w = 128; w > 0; w >>= 1) { if (tid < w) red[tid] = fmaxf(red[tid], red[tid + w]); __syncthreads(); }
  float hmax = red[0]; __syncthreads();
  for (int l = tid; l < Ln; l += 256) e[b * Ln + l] = expf(s1 * (height[b * Ln + l] - hmax));
  __syncthreads();
  if (tid == 0) {
    float acc = 0.f; pex[b * 256] = 0.f;
    for (int l = 0; l < Ln; ++l) { acc += e[b * Ln + l]; pex[b * 256 + l + 1] = acc; }
  }
  if (b == 0) for (int l = tid; l < Ln; l += 256) oidx[l] = (float)l;
}

// per (b,i): gamma -> reverse/forward cummin scans -> bml, bmr, block row
__global__ void k_scans(const float* __restrict__ height, const float* __restrict__ distf,
                        const float* __restrict__ scaler, float* __restrict__ bml,
                        float* __restrict__ bmr, float* __restrict__ oblock)
{
  __shared__ float g[Ln], s[Ln], bmlc[Ln], bmrc[Ln];
  const int row = blockIdx.x, tid = threadIdx.x;
  const int b = row / Ln, i = row % Ln;
  const float s0 = expf(scaler[0]);
  const float hi = height[b * Ln + i];
  for (int j = tid; j < Ln; j += 256)
    g[j] = 1.f / (1.f + expf((distf[b * Ln + j] - hi) * s0));   // sigmoid(-beta)
  __syncthreads();
  // reverse inclusive cummin of x[j] = (j<i ? g[j] : 1): scan the reversed array
  if (tid < Ln) { int jr = Ln - 1 - tid; s[tid] = (jr < i) ? g[jr] : 1.f; }
  __syncthreads();
  for (int off = 1; off < Ln; off <<= 1) {
    float v = 0.f;
    if (tid < Ln) { v = s[tid]; if (tid >= off) v = fminf(v, s[tid - off]); }
    __syncthreads();
    if (tid < Ln) s[tid] = v;
    __syncthreads();
  }
  if (tid < Ln) bmlc[Ln - 1 - tid] = s[tid];
  __syncthreads();
  // forward inclusive cummin of xr[j] = (j==0 ? 1 : (i<=j-1 ? g[j-1] : 1))
  if (tid < Ln) s[tid] = (tid == 0) ? 1.f : ((i <= tid - 1) ? g[tid - 1] : 1.f);
  __syncthreads();
  for (int off = 1; off < Ln; off <<= 1) {
    float v = 0.f;
    if (tid < Ln) { v = s[tid]; if (tid >= off) v = fminf(v, s[tid - off]); }
    __syncthreads();
    if (tid < Ln) s[tid] = v;
    __syncthreads();
  }
  if (tid < Ln) bmrc[tid] = s[tid];
  __syncthreads();
  if (tid < Ln) {
    int j = tid;
    long base = (long)row * Ln;
    bml[base + j] = (j <= i) ? (bmlc[j] - (j > 0 ? bmlc[j - 1] : 0.f)) : 0.f;
    bmr[base + j] = (j >= i) ? (bmrc[j] - (j < Ln - 1 ? bmrc[j + 1] : 0.f)) : 0.f;
    oblock[base + j] = (j <= i) ? bmlc[j] : bmrc[j];   // cumsum telescopes
  }
}

// block_p[b,l,:,:] = outer(bml[b,l,:], bmr[b,l,:]) — pure store bandwidth
__global__ void k_blockp(const float* __restrict__ bml, const float* __restrict__ bmr,
                         float* __restrict__ bp)
{
  __shared__ float a[Ln], c[Ln];
  const int row = blockIdx.x, tid = threadIdx.x;
  for (int j = tid; j < Ln; j += 256) { a[j] = bml[(long)row * Ln + j]; c[j] = bmr[(long)row * Ln + j]; }
  __syncthreads();
  float* o = bp + (long)row * Ln * Ln;
  for (int idx = tid; idx < Ln * Ln; idx += 256)
    o[idx] = a[idx / Ln] * c[idx % Ln];
}

// head_p[b,i,j,h]: closed-form interval softmax e[h]/(P[j+1]-P[i]); i>j -> 1/L
__global__ void k_headp(const float* __restrict__ e, const float* __restrict__ pex,
                        float* __restrict__ hp)
{
  __shared__ float es[Ln], ps[Ln + 1];
  const int row = blockIdx.x, tid = threadIdx.x;
  const int b = row / Ln, i = row % Ln;
  for (int h = tid; h < Ln; h += 256) es[h] = e[b * Ln + h];
  for (int k = tid; k <= Ln; k += 256) ps[k] = pex[b * 256 + k];
  __syncthreads();
  float* o = hp + (long)row * Ln * Ln;
  const float unif = 1.f / (float)Ln;
  for (int idx = tid; idx < Ln * Ln; idx += 256) {
    int j = idx / Ln, h = idx % Ln;
    float v;
    if (i > j) v = unif;
    else {
      float Z = ps[j + 1] - ps[i];
      if (Z <= 0.f) v = unif;                       // all-(-NEG) degenerate row
      else v = (h >= i && h <= j) ? es[h] / Z : 0.f;
    }
    o[idx] = v;
  }
}

// ---------------- launch ------------------------------------------------------
extern "C" void kernel_launch(void* const* d_in, const int* in_sizes, int n_in,
                              void* d_out, int out_size, void* d_ws, size_t ws_size,
                              hipStream_t stream) {
  (void)in_sizes; (void)n_in; (void)out_size; (void)ws_size;
  const float* h       = (const float*)d_in[0];
  const int*   ids     = (const int*)d_in[1];
  const float* conv_w  = (const float*)d_in[2];
  const float* conv_b  = (const float*)d_in[3];
  const float* dist_w  = (const float*)d_in[4];
  const float* dist_b  = (const float*)d_in[5];
  const float* h1_w    = (const float*)d_in[6];
  const float* h1_b    = (const float*)d_in[7];
  const float* h2_w    = (const float*)d_in[8];
  const float* h2_b    = (const float*)d_in[9];
  const float* scaler  = (const float*)d_in[10];

  float* out      = (float*)d_out;               // tuple concatenated flat
  float* o_dist   = out;                         // [4,224]
  float* o_height = out + 896;                   // [4,224]
  float* o_cib    = out + 1792;                  // [4,224,224]
  float* o_headp  = out + 202496L;               // [4,224,224,224]
  float* o_head   = out + 45160192L;             // [4,224,224]
  float* o_blockp = out + 45360896L;             // [4,224,224,224]
  float* o_block  = out + 90318592L;             // [4,224,224]
  float* o_idx    = out + 90519296L;             // [224]

  float* w = (float*)d_ws;
  float* wt0      = w; w += (long)Hn * Hn;
  float* wt1      = w; w += (long)Hn * Hn;
  float* hshift   = w; w += (long)Bn * Ln * Hn;
  float* hh_pre   = w; w += (long)Bn * Ln * Hn;
  float* dpre     = w; w += (long)Bn * Ln * Hn;
  float* bml      = w; w += (long)Bn * Ln * Ln;
  float* bmr      = w; w += (long)Bn * Ln * Ln;
  float* ev       = w; w += Bn * Ln;
  float* pex      = w; w += Bn * 256;
  float* draw     = w; w += Bn * Ln;
  unsigned* mbits = (unsigned*)w;

  k_init<<<1, 1, 0, stream>>>(mbits);
  k_trans<<<(Hn * Hn + 255) / 256, 256, 0, stream>>>(conv_w, wt0, wt1);
  k_shift<<<(Bn * Ln * Hn + 255) / 256, 256, 0, stream>>>(h, hshift);

  dim3 gA(Hn / 16, (Bn * Ln) / 16, 1);           // 64 x 56 tiles, one wave each
  k_gemm<false, false, false><<<gA, 32, 0, stream>>>(
      h, h1_w, nullptr, nullptr, hh_pre, Hn, Hn, Hn, Hn, 0, 0, 0);
  k_gemm<false, false, true><<<gA, 32, 0, stream>>>(
      h, wt0, hshift, wt1, dpre, Hn, Hn, Hn, Hn, 0, 0, 0);

  k_ln<<<Bn * Ln, 256, 0, stream>>>(hh_pre, h1_b, h2_w, h2_b, ids, o_height, 0, -NEGC);
  k_ln<<<Bn * Ln, 256, 0, stream>>>(dpre, conv_b, dist_w, dist_b, ids, draw, 1, NEGC);

  k_margin<<<7, 128, 0, stream>>>(o_height, draw, ids, mbits);
  k_finalize<<<Bn, 256, 0, stream>>>(o_height, draw, (const float*)mbits, scaler,
                                     o_dist, ev, pex, o_idx);
  k_scans<<<Bn * Ln, 256, 0, stream>>>(o_height, o_dist, scaler, bml, bmr, o_block);
  k_blockp<<<Bn * Ln, 256, 0, stream>>>(bml, bmr, o_blockp);
  k_headp<<<Bn * Ln, 256, 0, stream>>>(ev, pex, o_headp);

  // head[b] = block_p[b](224 x 50176) @ head_p[b](50176 x 224), diag zeroed
  dim3 gE(Ln / 16, Ln / 16, Bn);
  k_gemm<false, true, false><<<gE, 32, 0, stream>>>(
      o_blockp, o_headp, nullptr, nullptr, o_head,
      Ln * Ln, Ln * Ln, Ln, Ln,
      (long)Ln * Ln * Ln, (long)Ln * Ln * Ln, (long)Ln * Ln);
  // cibling[b] = head[b] @ head[b]^T, diag zeroed
  k_gemm<true, true, false><<<gE, 32, 0, stream>>>(
      o_head, o_head, nullptr, nullptr, o_cib,
      Ln, Ln, Ln, Ln,
      (long)Ln * Ln, (long)Ln * Ln, (long)Ln * Ln);
}